// ConvertParamsToMomentsLayer_76596446757537
// MI455X (gfx1250) — compile-verified
//
#include <hip/hip_runtime.h>
#include <hip/hip_bf16.h>
#include <stdint.h>

#define BATCH 256
#define NVDIM 512
#define NHDIM 256
#define NTOT  768   // NV + NH

typedef __attribute__((ext_vector_type(2))) float v2f;
typedef __attribute__((ext_vector_type(4))) float v4f;
typedef __attribute__((ext_vector_type(8))) float v8f;

// LDS geometry for the GEMM kernel (floats):
//   double-buffered operand stage: buf b at b*5120: [As 2560 | Bs 2560]
//   K-pair interleaved: element (k, col) at (k>>1)*160 + col*2 + (k&1)
//   (K-pair stride 160 => the two lane-halves of a fragment read hit
//    disjoint bank halves; a {K,K+1} v2f fragment is one ds_load_b64)
#define KP_STRIDE 160
#define BUF_FLOATS 5120
#define BS_OFF 2560

// ---------------------------------------------------------------------------
// Kernel 1: varv[b] = wt[b]^T * diag(varh[b]) * wt[b] + sig2[b]*I
// 128 threads = 4 waves per 64x64 output tile; 2x2 WMMA accums per wave.
// Global->LDS staging via async-to-LDS DMA (ASYNCcnt), double buffered.
// Output staged through LDS for b128 non-temporal stores.
// ---------------------------------------------------------------------------
__device__ __forceinline__ void issue_chunk_async(
    const float* __restrict__ gA,   // wt row-chunk base at column v0
    const float* __restrict__ gB,   // wt row-chunk base at column u0
    uint32_t ldsA, uint32_t ldsB, int t)
{
#pragma unroll
    for (int e = 0; e < 16; ++e) {
        const int idx  = e * 128 + t;       // 2048 floats per operand buffer
        const int k    = idx >> 6;          // 0..31 row within chunk
        const int col  = idx & 63;          // 0..63 column within tile
        const int goff = k * NVDIM + col;   // floats
        const uint32_t loff =
            (uint32_t)(((k >> 1) * KP_STRIDE + col * 2 + (k & 1)) * 4);
        asm volatile("global_load_async_to_lds_b32 %0, %1, off"
                     :: "v"(ldsA + loff), "v"(gA + goff) : "memory");
        asm volatile("global_load_async_to_lds_b32 %0, %1, off"
                     :: "v"(ldsB + loff), "v"(gB + goff) : "memory");
    }
}

__global__ __launch_bounds__(128) void varv_wmma_kernel(
    const float* __restrict__ wt,     // [B, NH, NV]
    const float* __restrict__ varh,   // [B, NH]
    const float* __restrict__ sig2,   // [B]
    float* __restrict__ var_out)      // [B, 768, 768]
{
    const int bidx = blockIdx.z;
    const int u0 = blockIdx.x * 64;   // output col tile
    const int v0 = blockIdx.y * 64;   // output row tile

    const float* __restrict__ wtb = wt   + (size_t)bidx * NHDIM * NVDIM;
    const float* __restrict__ db  = varh + (size_t)bidx * NHDIM;
    float* __restrict__ outb      = var_out + (size_t)bidx * NTOT * NTOT;

    __shared__ float smem[2 * BUF_FLOATS];   // 40 KB operand stage (+ store stage reuse)
    __shared__ float sd[NHDIM];              // varh_diag, folded into B fragments

    const int t    = threadIdx.x;
    const int lane = t & 31;
    const int wave = t >> 5;
    const int wy   = wave >> 1;
    const int wx   = wave & 1;
    const int m0   = wy * 32;
    const int n0   = wx * 32;
    const int ml   = lane & 15;       // row/col within 16 for fragments
    const int half = lane >> 4;       // 0: K={k,k+1}, 1: K={k+2,k+3}

    // preload diag once (1 KB, broadcast reads later)
    sd[t]       = db[t];
    sd[t + 128] = db[t + 128];

    const uint32_t ldsbase = (uint32_t)(uintptr_t)smem;
    const float* gA = wtb + v0;
    const float* gB = wtb + u0;

    // prologue: chunk 0 -> buffer 0
    issue_chunk_async(gA, gB, ldsbase, ldsbase + BS_OFF * 4, t);

    v8f acc[2][2] = {};

    for (int c = 0; c < 8; ++c) {
        const int cur = c & 1;
        if (c + 1 < 8) {
            const int nxt = cur ^ 1;
            const float* gAn = gA + (c + 1) * 32 * NVDIM;
            const float* gBn = gB + (c + 1) * 32 * NVDIM;
            const uint32_t lb = ldsbase + (uint32_t)(nxt * BUF_FLOATS * 4);
            issue_chunk_async(gAn, gBn, lb, lb + BS_OFF * 4, t);
            // 32 ops just issued for chunk c+1 may stay in flight; in-order
            // completion => chunk c's loads have landed.
            asm volatile("s_wait_asynccnt 0x20" ::: "memory");
        } else {
            asm volatile("s_wait_asynccnt 0x0" ::: "memory");
        }
        __syncthreads();   // all waves' async data + sd visible

        const float* As = smem + cur * BUF_FLOATS;
        const float* Bs = As + BS_OFF;
        const int h0 = c * 32;

#pragma unroll
        for (int kk = 0; kk < 32; kk += 4) {
            const int kp = (kk >> 1) + half;     // K-pair index for this lane half
            // diag pair for this lane's K values (broadcast ds_load_b64)
            const v2f dpair = *(const v2f*)&sd[h0 + kk + 2 * half];
            v2f afrag[2], bfrag[2];
#pragma unroll
            for (int i = 0; i < 2; ++i)
                afrag[i] = *(const v2f*)&As[kp * KP_STRIDE + (m0 + i * 16 + ml) * 2];
#pragma unroll
            for (int j = 0; j < 2; ++j) {
                bfrag[j] = *(const v2f*)&Bs[kp * KP_STRIDE + (n0 + j * 16 + ml) * 2];
                bfrag[j] = bfrag[j] * dpair;      // fold diag(varh) into B
            }
#pragma unroll
            for (int i = 0; i < 2; ++i)
#pragma unroll
                for (int j = 0; j < 2; ++j)
                    acc[i][j] = __builtin_amdgcn_wmma_f32_16x16x4_f32(
                        false, afrag[i], false, bfrag[j],
                        (short)0, acc[i][j], false, false);
        }
        __syncthreads();   // everyone done with buf[cur] before it is refilled
    }

    // ---- store: stage tile in LDS (row stride 68), then b128 NT stores ----
    const float s2 = sig2[bidx];
#pragma unroll
    for (int i = 0; i < 2; ++i) {
#pragma unroll
        for (int j = 0; j < 2; ++j) {
            const int col_l = n0 + j * 16 + ml;
#pragma unroll
            for (int r = 0; r < 8; ++r) {
                const int row_l = m0 + i * 16 + half * 8 + r;
                float val = acc[i][j][r];
                if (v0 + row_l == u0 + col_l) val += s2;   // sigma^2 on diagonal
                smem[row_l * 68 + col_l] = val;
            }
        }
    }
    __syncthreads();
#pragma unroll
    for (int e = 0; e < 8; ++e) {
        const int idx = e * 128 + t;       // 64 rows x 16 v4 per row
        const int row = idx >> 4;
        const int c4  = (idx & 15) << 2;
        const v4f val = *(const v4f*)&smem[row * 68 + c4];
        __builtin_nontemporal_store(
            val, (v4f*)(outb + (size_t)(v0 + row) * NTOT + u0 + c4));
    }
}

// ---------------------------------------------------------------------------
// Kernel 2: pure-bandwidth remainder. One 256-thread block per batch.
//   mu[b]  = [ b + wt^T muh , muh ]
//   var[b] bottom-left  (512+h, v):      d[h]*wt[h,v]
//   var[b] top-right    (v, 512+h):      d[h]*wt[h,v]
//   var[b] bottom-right (512+h, 512+h'): diag(d)
// ---------------------------------------------------------------------------
__global__ __launch_bounds__(256) void fill_rest_kernel(
    const float* __restrict__ wt,
    const float* __restrict__ varh,
    const float* __restrict__ bvec,
    const float* __restrict__ muh,
    float* __restrict__ mu_out,
    float* __restrict__ var_out)
{
    const int bidx = blockIdx.x;
    const int t = threadIdx.x;
    const float* __restrict__ wtb = wt + (size_t)bidx * NHDIM * NVDIM;
    float* __restrict__ mub  = mu_out  + (size_t)bidx * NTOT;
    float* __restrict__ varb = var_out + (size_t)bidx * NTOT * NTOT;

    __shared__ float sd[NHDIM];
    __shared__ float smuh[NHDIM];
    sd[t]   = varh[(size_t)bidx * NHDIM + t];
    smuh[t] = muh[(size_t)bidx * NHDIM + t];
    __syncthreads();

    // (a) mu: two coalesced column-dot-products per thread
    {
        float acc0 = 0.f, acc1 = 0.f;
        for (int h = 0; h < NHDIM; ++h) {
            const float m = smuh[h];
            const float* rowp = wtb + (size_t)h * NVDIM;
            acc0 = __builtin_fmaf(rowp[t], m, acc0);
            acc1 = __builtin_fmaf(rowp[256 + t], m, acc1);
        }
        mub[t]         = bvec[(size_t)bidx * NVDIM + t] + acc0;
        mub[256 + t]   = bvec[(size_t)bidx * NVDIM + 256 + t] + acc1;
        mub[NVDIM + t] = smuh[t];
    }

    // (b) bottom-left: var[512+h][v] = d[h]*wt[h][v]
    for (int it = 0; it < 128; ++it) {
        const int idx = it * 256 + t;
        const int h = idx >> 7;
        const int c = (idx & 127) << 2;
        v4f w4 = *(const v4f*)(wtb + (size_t)h * NVDIM + c);
        w4 = w4 * sd[h];
        __builtin_nontemporal_store(
            w4, (v4f*)(varb + (size_t)(NVDIM + h) * NTOT + c));
    }

    // (c) top-right: var[v][512+h] = d[h]*wt[h][v]  (L2-resident column reads)
    {
        const int v_off = t >> 6;
        const int h4 = (t & 63) << 2;
        for (int vb = 0; vb < NVDIM; vb += 4) {
            const int v = vb + v_off;
            v4f val;
            val.x = sd[h4 + 0] * wtb[(size_t)(h4 + 0) * NVDIM + v];
            val.y = sd[h4 + 1] * wtb[(size_t)(h4 + 1) * NVDIM + v];
            val.z = sd[h4 + 2] * wtb[(size_t)(h4 + 2) * NVDIM + v];
            val.w = sd[h4 + 3] * wtb[(size_t)(h4 + 3) * NVDIM + v];
            __builtin_nontemporal_store(
                val, (v4f*)(varb + (size_t)v * NTOT + NVDIM + h4));
        }
    }

    // (d) bottom-right: diag(varh) block
    for (int it = 0; it < 64; ++it) {
        const int idx = it * 256 + t;
        const int h = idx >> 6;
        const int c4 = (idx & 63) << 2;
        v4f val = {0.f, 0.f, 0.f, 0.f};
        if (h >= c4 && h < c4 + 4) val[h - c4] = sd[h];
        __builtin_nontemporal_store(
            val, (v4f*)(varb + (size_t)(NVDIM + h) * NTOT + NVDIM + c4));
    }
}

extern "C" void kernel_launch(void* const* d_in, const int* in_sizes, int n_in,
                              void* d_out, int out_size, void* d_ws, size_t ws_size,
                              hipStream_t stream) {
    const float* wt   = (const float*)d_in[0];  // [B, NH, NV]
    const float* varh = (const float*)d_in[1];  // [B, NH]
    const float* sig2 = (const float*)d_in[2];  // [B]
    const float* bvec = (const float*)d_in[3];  // [B, NV]
    const float* muh  = (const float*)d_in[4];  // [B, NH]

    float* mu_out  = (float*)d_out;                       // [B, 768]
    float* var_out = mu_out + (size_t)BATCH * NTOT;       // [B, 768, 768]

    dim3 g1(NVDIM / 64, NVDIM / 64, BATCH);               // (8, 8, 256)
    varv_wmma_kernel<<<g1, 128, 0, stream>>>(wt, varh, sig2, var_out);
    fill_rest_kernel<<<BATCH, 256, 0, stream>>>(wt, varh, bvec, muh, mu_out, var_out);
}